// DynamicCausalMemoryNetwork_42348377538863
// MI455X (gfx1250) — compile-verified
//
#include <hip/hip_runtime.h>
#include <math.h>

typedef float v2f __attribute__((ext_vector_type(2)));
typedef float v4f __attribute__((ext_vector_type(4)));
typedef float v8f __attribute__((ext_vector_type(8)));
typedef int   vi4 __attribute__((vector_size(16)));   // matches builtin param type

#define B_       64
#define N_       1024
#define S_LOC    12
#define HID      64
#define PAT      32
#define M_GLB    64
#define S_GLB    6
#define M_LOC    64
#define NEMB     16
#define BN       (B_ * N_)       // 65536
#define KDIM     (S_LOC * HID)   // 768
#define MCOLS    128             // M_LOC + M_GLB

// ---------------------------------------------------------------------------
// CDNA5 async global->LDS copy (ASYNCcnt-tracked), with portable fallback.
// Builtin signature (from probe diagnostics): (AS1 int4*, AS3 int4*, Ii, Ii).
// ---------------------------------------------------------------------------
#if __has_builtin(__builtin_amdgcn_global_load_async_to_lds_b128)
#define HAVE_ASYNC_LDS 1
__device__ __forceinline__ void async_g2l_b128(const float* g, float* l) {
    __builtin_amdgcn_global_load_async_to_lds_b128(
        (__attribute__((address_space(1))) vi4*)(uintptr_t)g,
        (__attribute__((address_space(3))) vi4*)(uint32_t)(uintptr_t)l,
        0, 0);
}
__device__ __forceinline__ void wait_async_lds() {
#if __has_builtin(__builtin_amdgcn_s_wait_asynccnt)
    __builtin_amdgcn_s_wait_asynccnt(0);
#else
    asm volatile("s_wait_asynccnt 0x0" ::: "memory");
#endif
}
#else
#define HAVE_ASYNC_LDS 0
__device__ __forceinline__ void async_g2l_b128(const float* g, float* l) {
    *(v4f*)l = *(const v4f*)g;
}
__device__ __forceinline__ void wait_async_lds() {}
#endif

// ---------------------------------------------------------------------------
// Kernel 1: fold W_q/b_q into the memory banks.
//   Bt[col][k]   (col-major, 128 x 768): B-fragments are contiguous b64 loads
//   cvec[128]    fused bias
//   LmeanT/GmeanT [32][64]: transposed bank means (B-operand layout for the
//   context-retrieval WMMA in kernel 2's epilogue)
// ---------------------------------------------------------------------------
__global__ __launch_bounds__(256)
void precompute_kernel(const float* __restrict__ L,   // [64][12][32]
                       const float* __restrict__ G,   // [64][6][32]
                       const float* __restrict__ Wq,  // [64][32]
                       const float* __restrict__ bq,  // [32]
                       float* __restrict__ Bt,        // [128][768]
                       float* __restrict__ cvec,      // [128]
                       float* __restrict__ LmeanT,    // [32][64]
                       float* __restrict__ GmeanT)    // [32][64]
{
    const int m = blockIdx.x;              // 0..127
    const int t = threadIdx.x;             // 0..255
    const bool isG = (m >= M_LOC);
    const int mm = isG ? (m - M_LOC) : m;

    for (int j = t; j < KDIM; j += 256) {
        const int s = j >> 6;              // 0..11
        const int h = j & 63;
        float acc = 0.f;
        if (!isG) {
            const float* lrow = L + (mm * S_LOC + s) * PAT;
            #pragma unroll
            for (int p = 0; p < PAT; ++p) acc += Wq[h * PAT + p] * lrow[p];
        } else {
            const float* grow = G + (mm * S_GLB + (s >> 1)) * PAT;
            #pragma unroll
            for (int p = 0; p < PAT; ++p) acc += Wq[h * PAT + p] * grow[p];
            acc *= 0.5f;   // avg-pool of q commutes into the pattern matrix
        }
        Bt[(size_t)m * KDIM + j] = acc;
    }

    if (t == 0) {
        float acc = 0.f;
        if (!isG) {
            for (int s = 0; s < S_LOC; ++s)
                for (int p = 0; p < PAT; ++p)
                    acc += bq[p] * L[(mm * S_LOC + s) * PAT + p];
        } else {
            for (int sg = 0; sg < S_GLB; ++sg)
                for (int p = 0; p < PAT; ++p)
                    acc += bq[p] * G[(mm * S_GLB + sg) * PAT + p];
        }
        cvec[m] = acc;
    }

    if (t < PAT) {
        if (!isG) {
            float s = 0.f;
            for (int i = 0; i < S_LOC; ++i) s += L[(mm * S_LOC + i) * PAT + t];
            LmeanT[t * 64 + mm] = s * (1.f / S_LOC);
        } else {
            float s = 0.f;
            for (int i = 0; i < S_GLB; ++i) s += G[(mm * S_GLB + i) * PAT + t];
            GmeanT[t * 64 + mm] = s * (1.f / S_GLB);
        }
    }
}

// ---------------------------------------------------------------------------
// Kernel 2: sim = hh[BN,768] @ Bt^T + c  via v_wmma_f32_16x16x4_f32,
// then dual softmax (VALU) + context retrieval (WMMA) -> fused[BN,64].
// 256 threads = 8 wave32.
// ---------------------------------------------------------------------------
__global__ __launch_bounds__(256)
void sim_softmax_ctx_kernel(const float* __restrict__ hh,     // [BN][768]
                            const float* __restrict__ Bt,     // [128][768]
                            const float* __restrict__ cvec,   // [128]
                            const float* __restrict__ LmeanT, // [32][64]
                            const float* __restrict__ GmeanT, // [32][64]
                            float* __restrict__ fused)        // [BN][64]
{
    __shared__ __align__(16) float lds_hh[16 * KDIM];        // 48 KB
    __shared__ __align__(16) float lds_sim[16 * MCOLS];      //  8 KB
    __shared__ __align__(16) float lds_meanT[2][PAT * 64];   // 16 KB
    __shared__ float lds_inv[16][2];

    const int t = threadIdx.x;
    const int rowBase = blockIdx.x * 16;

    // Stage hh tile: 16 consecutive rows = one contiguous 48KB region.
    {
        const float* src = hh + (size_t)rowBase * KDIM;
        for (int i = t * 4; i < 16 * KDIM; i += 256 * 4)
            async_g2l_b128(src + i, lds_hh + i);
    }
    for (int i = t; i < PAT * 64; i += 256) {
        lds_meanT[0][i] = LmeanT[i];
        lds_meanT[1][i] = GmeanT[i];
    }
    wait_async_lds();
    __syncthreads();

    const int lane = t & 31;
    const int wave = t >> 5;           // 0..7 -> column tile of sim
    const int ln   = lane & 15;
    const int koff = (lane >> 4) * 2;  // half-wave selects K {0,1} vs {2,3}
    const int colg = wave * 16 + ln;   // 0..127

    // ---- main GEMM: 16 x 128 sim tile, K = 768 --------------------------
    {
        v8f acc = {0.f, 0.f, 0.f, 0.f, 0.f, 0.f, 0.f, 0.f};
        const float* arow  = lds_hh + (size_t)ln * KDIM;
        const float* btcol = Bt + (size_t)colg * KDIM;
        for (int k0 = 0; k0 < KDIM; k0 += 4) {
            v2f a = *(const v2f*)(arow + k0 + koff);    // ds_load_b64
            v2f b = *(const v2f*)(btcol + k0 + koff);   // global_load_b64 (L2-hot)
            acc = __builtin_amdgcn_wmma_f32_16x16x4_f32(
                      false, a, false, b, (short)0, acc, false, false);
        }
        const float cc = cvec[colg];
        #pragma unroll
        for (int j = 0; j < 8; ++j) {
            const int r = j + (lane >> 4) * 8;          // C layout: VGPR j -> M=j / j+8
            lds_sim[r * MCOLS + colg] = acc[j] + cc;
        }
    }
    __syncthreads();

    // ---- softmax statistics: 32 tasks = 16 rows x {local, global} -------
    // Writes UNNORMALIZED exp weights back into lds_sim + 1/sum per task.
    if (t < 32) {
        const int r = t >> 1;
        const int half = t & 1;
        float* srow = lds_sim + r * MCOLS + half * 64;
        float mx = -1e30f;
        for (int m = 0; m < 64; ++m) mx = fmaxf(mx, srow[m]);
        float se = 0.f;
        for (int m = 0; m < 64; ++m) {
            const float e = __expf(srow[m] - mx);
            se += e;
            srow[m] = e;
        }
        lds_inv[r][half] = 1.f / se;
    }
    __syncthreads();

    // ---- context retrieval on matrix pipes:
    // fused_tile[16, 64] = attn[16, 2x64] @ meanT^T, 4 waves x 1 tile each.
    if (wave < 4) {
        const int half = wave >> 1;    // local / global
        const int pt   = wave & 1;     // which 16-wide slice of the 32 patterns
        v8f acc = {0.f, 0.f, 0.f, 0.f, 0.f, 0.f, 0.f, 0.f};
        const float* arow = lds_sim + ln * MCOLS + half * 64;
        const float* bcol = &lds_meanT[half][(pt * 16 + ln) * 64];
        #pragma unroll
        for (int k0 = 0; k0 < 64; k0 += 4) {
            v2f a = *(const v2f*)(arow + k0 + koff);
            v2f b = *(const v2f*)(bcol + k0 + koff);
            acc = __builtin_amdgcn_wmma_f32_16x16x4_f32(
                      false, a, false, b, (short)0, acc, false, false);
        }
        #pragma unroll
        for (int j = 0; j < 8; ++j) {
            const int r = j + (lane >> 4) * 8;
            fused[(size_t)(rowBase + r) * 64 + half * 32 + pt * 16 + ln] =
                acc[j] * lds_inv[r][half];
        }
    }
}

// ---------------------------------------------------------------------------
// Kernel 3: per node n: node_w = emb[n] @ pool  ([64f][64h]),
// out[:, n, :] = fused[:, n, :] @ node_w  (64x64 @ 64x64) via f32 WMMA.
// 128 threads = 4 wave32; wave w does rows [16w,16w+16) x all 4 col tiles.
// ---------------------------------------------------------------------------
__global__ __launch_bounds__(128)
void output_kernel(const float* __restrict__ fused,  // [BN][64], row = b*N + n
                   const float* __restrict__ nemb,   // [N][16]
                   const float* __restrict__ wpool,  // [16][64][64]
                   float* __restrict__ out)          // [B][N][64]
{
    __shared__ __align__(16) float nodewT[64 * 64];  // [h][f] (transposed) 16 KB
    __shared__ __align__(16) float fusedL[64 * 64];  // [b][f]             16 KB

    const int n = blockIdx.x;
    const int t = threadIdx.x;

    // Stage fused block asynchronously: fusedL[b][f] = fused[(b*N+n)*64 + f].
    for (int i4 = t; i4 < (64 * 64) / 4; i4 += 128) {
        const int b = i4 >> 4, f4 = (i4 & 15) * 4;
        async_g2l_b128(fused + ((size_t)b * N_ + n) * 64 + f4, fusedL + b * 64 + f4);
    }

    float e[NEMB];
    #pragma unroll
    for (int d = 0; d < NEMB; ++d) e[d] = nemb[n * NEMB + d];

    // nodewT[h][f] = sum_d e[d] * wpool[d][f][h]; lanes walk h -> coalesced.
    {
        const int h  = t & 63;
        const int f0 = (t >> 6) * 32;
        for (int f = f0; f < f0 + 32; ++f) {
            float acc = 0.f;
            #pragma unroll
            for (int d = 0; d < NEMB; ++d)
                acc += e[d] * wpool[((size_t)d * 64 + f) * 64 + h];
            nodewT[h * 64 + f] = acc;
        }
    }
    wait_async_lds();
    __syncthreads();

    const int lane = t & 31;
    const int wave = t >> 5;
    const int ln   = lane & 15;
    const int koff = (lane >> 4) * 2;
    const int rowB = wave * 16;

    v8f acc[4];
    #pragma unroll
    for (int ct = 0; ct < 4; ++ct) acc[ct] = (v8f){0.f,0.f,0.f,0.f,0.f,0.f,0.f,0.f};

    for (int k0 = 0; k0 < 64; k0 += 4) {
        v2f a = *(const v2f*)(&fusedL[(rowB + ln) * 64 + k0 + koff]);
        #pragma unroll
        for (int ct = 0; ct < 4; ++ct) {
            v2f b = *(const v2f*)(&nodewT[(ct * 16 + ln) * 64 + k0 + koff]);
            acc[ct] = __builtin_amdgcn_wmma_f32_16x16x4_f32(
                          false, a, false, b, (short)0, acc[ct], false, false);
        }
    }

    #pragma unroll
    for (int ct = 0; ct < 4; ++ct) {
        #pragma unroll
        for (int j = 0; j < 8; ++j) {
            const int b = rowB + j + (lane >> 4) * 8;
            const int h = ct * 16 + ln;
            out[((size_t)b * N_ + n) * 64 + h] = acc[ct][j];
        }
    }
}

// ---------------------------------------------------------------------------
extern "C" void kernel_launch(void* const* d_in, const int* in_sizes, int n_in,
                              void* d_out, int out_size, void* d_ws, size_t ws_size,
                              hipStream_t stream) {
    const float* hh    = (const float*)d_in[0];  // h_history [64,1024,12,64]
    const float* G     = (const float*)d_in[1];  // global_memory_bank [64,6,32]
    const float* L     = (const float*)d_in[2];  // local_memory_bank [64,12,32]
    const float* Wq    = (const float*)d_in[3];  // [64,32]
    const float* bq    = (const float*)d_in[4];  // [32]
    const float* nemb  = (const float*)d_in[5];  // [1024,16]
    const float* wpool = (const float*)d_in[6];  // [16,64,64]
    float* out = (float*)d_out;

    float* Bt     = (float*)d_ws;                  // 128*768
    float* cvec   = Bt + (size_t)MCOLS * KDIM;     // 128
    float* LmeanT = cvec + MCOLS;                  // 32*64
    float* GmeanT = LmeanT + PAT * 64;             // 32*64
    float* fused  = GmeanT + PAT * 64;             // 65536*64 (~16 MB)

    precompute_kernel<<<MCOLS, 256, 0, stream>>>(L, G, Wq, bq, Bt, cvec, LmeanT, GmeanT);
    sim_softmax_ctx_kernel<<<BN / 16, 256, 0, stream>>>(hh, Bt, cvec, LmeanT, GmeanT, fused);
    output_kernel<<<N_, 128, 0, stream>>>(fused, nemb, wpool, out);
}